// AdaptiveSimpleGCNConv_32641751449979
// MI455X (gfx1250) — compile-verified
//
#include <hip/hip_runtime.h>

// GCN conv: out = norm(A+I) @ x @ W^T + b, exploiting sparsity (E=262144 << N^2)
// and CDNA5 V_WMMA_F32_16X16X4_F32 for the final fp32 linear layer.

#define N_NODES 8192
#define D_FEAT  128
#define ROW_WORDS (N_NODES / 32)                 // 256 words per adjacency row
#define MASK_WORDS (N_NODES * ROW_WORDS)         // 2,097,152 words = 8 MB

typedef __attribute__((ext_vector_type(2))) float v2f;
typedef __attribute__((ext_vector_type(8))) float v8f;

// ---------------------------------------------------------------- zero scratch
__global__ void gcn_zero_kernel(unsigned int* __restrict__ p, int n) {
    int i = blockIdx.x * blockDim.x + threadIdx.x;
    if (i < n) p[i] = 0u;
}

// ------------------------------------------- dedup edges into bitmask + degree
// Duplicates collapse (atomicOr old-value says who owns the bit). Integer
// atomics only -> deterministic regardless of arrival order.
__global__ void gcn_edge_kernel(const long long* __restrict__ e_row,
                                const long long* __restrict__ e_col,
                                unsigned int* __restrict__ mask,
                                unsigned int* __restrict__ deg, int E) {
    int i = blockIdx.x * blockDim.x + threadIdx.x;
    if (i >= E) return;
    unsigned int r = (unsigned int)e_row[i];
    unsigned int c = (unsigned int)e_col[i];
    unsigned int idx = r * (unsigned int)N_NODES + c;     // < 2^26, fits u32
    unsigned int bit = 1u << (idx & 31u);
    unsigned int old = atomicOr(&mask[idx >> 5], bit);
    if (!(old & bit)) atomicAdd(&deg[r], 1u);
}

// ------------------------------------------------- degree -> D^{-1/2} (exact)
// degree = popcount(row bits) + 1 (the +I term); always > 0.
__global__ void gcn_dinv_kernel(const unsigned int* __restrict__ deg,
                                float* __restrict__ dinv) {
    int i = blockIdx.x * blockDim.x + threadIdx.x;
    if (i < N_NODES) dinv[i] = 1.0f / sqrtf((float)(deg[i] + 1u));
}

// ----------------------------- sparse aggregation: agg = D^-1/2 (A+I) D^-1/2 x
// One block (128 threads) per row; thread t owns feature dim t. Row bitmask
// staged in LDS; fixed set-bit iteration order -> deterministic fp sums.
__global__ void gcn_agg_kernel(const float* __restrict__ x,
                               const unsigned int* __restrict__ mask,
                               const float* __restrict__ dinv,
                               float* __restrict__ agg) {
    int row = blockIdx.x;
    int t   = threadIdx.x;                       // 0..127
    __shared__ unsigned int mrow[ROW_WORDS];
    const unsigned int* m = mask + (size_t)row * ROW_WORDS;
    mrow[t]       = m[t];
    mrow[t + 128] = m[t + 128];
    __syncthreads();

    float di = dinv[row];
    // +I contributes one extra dinv[row]*x[row] term on top of any self-edge bit.
    float s = di * x[(size_t)row * D_FEAT + t];
    for (int w = 0; w < ROW_WORDS; ++w) {
        unsigned int bits = mrow[w];
        int base = w << 5;
        while (bits) {
            int j = base + __builtin_ctz(bits);
            bits &= bits - 1u;
            s = fmaf(dinv[j], x[(size_t)j * D_FEAT + t], s);
        }
    }
    agg[(size_t)row * D_FEAT + t] = di * s;
}

// -------------------------------- fp32 WMMA GEMM: out = agg @ W^T + b
// One wave per 16x16 output tile; 32x V_WMMA_F32_16X16X4_F32 along K=128.
// A frag (16x4 f32): lanes 0-15 hold K=k,k+1; lanes 16-31 hold K=k+2,k+3
//   for row M = lane%16  -> contiguous float2 load from agg.
// B frag (4x16 f32):  column N = lane%16; lane-half selects K offset 0 / 2.
//   B[k'][n] = W[n][k'] -> contiguous float2 load from W row n.
// C/D (16x16 f32, 8 VGPRs): col = lane%16, row = v + 8*(lane/16).
__global__ void gcn_linear_wmma_kernel(const float* __restrict__ A,
                                       const float* __restrict__ W,
                                       const float* __restrict__ bias,
                                       float* __restrict__ out) {
    int tileM = blockIdx.x;          // 0..511
    int tileN = blockIdx.y;          // 0..7
    int lane  = threadIdx.x;         // 0..31
    int half  = lane >> 4;           // 0 or 1
    int l     = lane & 15;

    int arow = tileM * 16 + l;       // A row handled by this lane
    int bcol = tileN * 16 + l;       // output column handled by this lane

    v8f acc = {};
    #pragma unroll 4
    for (int k = 0; k < D_FEAT; k += 4) {
        int ka = k + half * 2;
        v2f a = *(const v2f*)(A + (size_t)arow * D_FEAT + ka);
        v2f b = *(const v2f*)(W + (size_t)bcol * D_FEAT + ka);
        acc = __builtin_amdgcn_wmma_f32_16x16x4_f32(
            /*neg_a=*/false, a, /*neg_b=*/false, b,
            /*c_mod=*/(short)0, acc, /*reuse_a=*/false, /*reuse_b=*/false);
    }

    float bv = bias[bcol];
    int rowBase = tileM * 16 + half * 8;
    #pragma unroll
    for (int v = 0; v < 8; ++v) {
        out[(size_t)(rowBase + v) * D_FEAT + bcol] = acc[v] + bv;
    }
}

// ---------------------------------------------------------------------- launch
extern "C" void kernel_launch(void* const* d_in, const int* in_sizes, int n_in,
                              void* d_out, int out_size, void* d_ws, size_t ws_size,
                              hipStream_t stream) {
    const float*     x  = (const float*)d_in[0];
    const long long* ei = (const long long*)d_in[1];   // int64 [2, E] flat
    const float*     W  = (const float*)d_in[2];
    const float*     b  = (const float*)d_in[3];
    float*           out = (float*)d_out;
    const int E = in_sizes[1] / 2;

    // Workspace layout: mask (8MB) | deg (32KB) | dinv (32KB) | agg (4MB)
    unsigned char* ws = (unsigned char*)d_ws;
    unsigned int* mask = (unsigned int*)ws;
    unsigned int* deg  = (unsigned int*)(ws + (size_t)MASK_WORDS * 4);
    float*        dinv = (float*)(ws + (size_t)MASK_WORDS * 4 + (size_t)N_NODES * 4);
    float*        agg  = (float*)(ws + (size_t)MASK_WORDS * 4 + (size_t)N_NODES * 8);

    // deg[] is contiguous after mask[], so one zero pass covers both.
    int zwords = MASK_WORDS + N_NODES;
    gcn_zero_kernel<<<(zwords + 255) / 256, 256, 0, stream>>>(mask, zwords);

    gcn_edge_kernel<<<(E + 255) / 256, 256, 0, stream>>>(ei, ei + E, mask, deg, E);

    gcn_dinv_kernel<<<(N_NODES + 255) / 256, 256, 0, stream>>>(deg, dinv);

    gcn_agg_kernel<<<N_NODES, 128, 0, stream>>>(x, mask, dinv, agg);

    dim3 grid(N_NODES / 16, D_FEAT / 16);
    gcn_linear_wmma_kernel<<<grid, 32, 0, stream>>>(agg, W, b, out);
}